// GatedGCNLayer_46102178955281
// MI455X (gfx1250) — compile-verified
//
#include <hip/hip_runtime.h>

typedef float v2f __attribute__((ext_vector_type(2)));
typedef float v4f __attribute__((ext_vector_type(4)));
typedef float v8f __attribute__((ext_vector_type(8)));

#define N_NODES 50000
#define N_EDGES 600000
#define DIM 128
#define LDSROW 132                 // pad 128 -> 132 dwords (row stride == 4 mod 64 banks)
#define TILE_FLOATS (16 * LDSROW)  // one 16x128 tile (padded)
#define TILE_BYTES (TILE_FLOATS * 4)
#define EPS_F 1e-6f

__device__ __forceinline__ v8f wmma_f32_k4(v2f a, v2f b, v8f c) {
  // V_WMMA_F32_16X16X4_F32: D = A(16x4) x B(4x16) + C(16x16), fp32 throughout.
  return __builtin_amdgcn_wmma_f32_16x16x4_f32(
      false, a, false, b, (short)0, c, false, false);
}

// Async DMA 32 bytes/lane global -> LDS (two B128 ops, ASYNCcnt-tracked).
// lds_byte is the LDS byte offset (flat shared addr truncated to 32 bits).
__device__ __forceinline__ void async_stage32(unsigned lds_byte, const float* gsrc) {
  asm volatile(
      "global_load_async_to_lds_b128 %0, %1, off\n\t"
      "global_load_async_to_lds_b128 %0, %1, off offset:16"
      :: "v"(lds_byte), "v"(gsrc) : "memory");
}
__device__ __forceinline__ void wait_async0() {
  asm volatile("s_wait_asynccnt 0" ::: "memory");
}

// ---------------------------------------------------------------------------
// Zero the two segment-sum buffers (graph-safe, no hipMemset).
// ---------------------------------------------------------------------------
__global__ void k_zero(float* __restrict__ p, int n4) {
  int i = blockIdx.x * blockDim.x + threadIdx.x;
  if (i < n4) ((v4f*)p)[i] = (v4f){0.f, 0.f, 0.f, 0.f};
}

// ---------------------------------------------------------------------------
// Node GEMMs: out = h @ W.T + b for W in {WA, WB, WD, WE}.
// blockIdx.x selects the matrix; wave w (of 8) owns output cols [16w,16w+16).
// B-fragments (full K=128) hoisted once per wave; 16-row h tiles are
// double-buffered in LDS via async global->LDS DMA.
// ---------------------------------------------------------------------------
__global__ void __launch_bounds__(256) k_node_gemm(
    const float* __restrict__ h,
    const float* __restrict__ WA, const float* __restrict__ bA,
    const float* __restrict__ WB, const float* __restrict__ bB,
    const float* __restrict__ WD, const float* __restrict__ bD,
    const float* __restrict__ WE, const float* __restrict__ bE,
    float* __restrict__ ws) {
  __shared__ float lds[2 * TILE_FLOATS];
  const int tid  = threadIdx.x;
  const int wave = tid >> 5;
  const int lane = tid & 31;
  const int nl   = lane & 15;          // col-within-tile / row M for A frag
  const int kk   = (lane >> 4) << 1;   // K sub-offset (0 or 2)
  const int colBase = wave * 16;

  const float* W; const float* bias; float* out;
  const size_t ND = (size_t)N_NODES * DIM;
  const int m = blockIdx.x;
  if (m == 0)      { W = WA; bias = bA; out = ws + 0 * ND; }
  else if (m == 1) { W = WB; bias = bB; out = ws + 1 * ND; }
  else if (m == 2) { W = WD; bias = bD; out = ws + 2 * ND; }
  else             { W = WE; bias = bE; out = ws + 3 * ND; }

  // B fragment (4x16 K x N): VGPR0 = {K=k0 | K=k0+2}, VGPR1 = {k0+1 | k0+3};
  // B[k][n] = W[n][k].
  v2f Bfrag[32];
  {
    const float* wrow = W + (size_t)(colBase + nl) * DIM + kk;
#pragma unroll
    for (int ks = 0; ks < 32; ++ks) Bfrag[ks] = *(const v2f*)(wrow + 4 * ks);
  }
  const float bv = bias[colBase + nl];

  const int ldrow = tid >> 4;          // staging: row 0..15
  const int ldcol = (tid & 15) * 8;    // 8 floats (32 B) per thread
  const unsigned ldsStage =
      (unsigned)(size_t)&lds[(size_t)ldrow * LDSROW + ldcol];

  const int NT = N_NODES / 16;
  const int step = gridDim.y;
  int t = blockIdx.y;
  int cur = 0;

  if (t < NT)
    async_stage32(ldsStage, h + (size_t)(t * 16 + ldrow) * DIM + ldcol);
  wait_async0();
  __syncthreads();

  for (; t < NT; t += step) {
    const int tn = t + step;
    if (tn < NT)  // prefetch next tile into the other buffer (overlaps compute)
      async_stage32(ldsStage + (cur ^ 1) * TILE_BYTES,
                    h + (size_t)(tn * 16 + ldrow) * DIM + ldcol);

    v8f acc0 = {bv, bv, bv, bv, bv, bv, bv, bv};
    v8f acc1 = {0.f, 0.f, 0.f, 0.f, 0.f, 0.f, 0.f, 0.f};
    const float* arow = &lds[cur * TILE_FLOATS + nl * LDSROW + kk];
#pragma unroll
    for (int ks = 0; ks < 32; ks += 2) {  // two chains hide WMMA latency
      v2f a0 = *(const v2f*)(arow + 4 * ks);
      acc0 = wmma_f32_k4(a0, Bfrag[ks], acc0);
      v2f a1 = *(const v2f*)(arow + 4 * (ks + 1));
      acc1 = wmma_f32_k4(a1, Bfrag[ks + 1], acc1);
    }

    // C/D layout: VGPR j -> row (j + 8*(lane>>4)), col (colBase + nl)
    const int r0 = (lane >> 4) * 8;
    float* orow = out + (size_t)(t * 16 + r0) * DIM + colBase + nl;
#pragma unroll
    for (int j = 0; j < 8; ++j) orow[(size_t)j * DIM] = acc0[j] + acc1[j];

    wait_async0();
    __syncthreads();
    cur ^= 1;
  }
}

// ---------------------------------------------------------------------------
// Fused edge kernel: Ce tile via WMMA (e tile double-buffered in LDS), then
//   e_new = Ce + Dh[src] + Eh[dst];  sigma = sigmoid(e_new)
//   e_out = e_in + e_new * sigma     (silu)
//   atomic segment sums: sum_h[dst] += Bh[src]*sigma ; sum_s[dst] += sigma
// ---------------------------------------------------------------------------
__global__ void __launch_bounds__(256) k_edge(
    const float* __restrict__ e,
    const int* __restrict__ src, const int* __restrict__ dst,
    const float* __restrict__ WC, const float* __restrict__ bC,
    const float* __restrict__ Bh, const float* __restrict__ Dh,
    const float* __restrict__ Eh,
    float* __restrict__ sum_h, float* __restrict__ sum_s,
    float* __restrict__ e_out) {
  __shared__ float lds[2 * TILE_FLOATS];
  const int tid  = threadIdx.x;
  const int wave = tid >> 5;
  const int lane = tid & 31;
  const int nl   = lane & 15;
  const int kk   = (lane >> 4) << 1;
  const int colBase = wave * 16;

  v2f Bfrag[32];
  {
    const float* wrow = WC + (size_t)(colBase + nl) * DIM + kk;
#pragma unroll
    for (int ks = 0; ks < 32; ++ks) Bfrag[ks] = *(const v2f*)(wrow + 4 * ks);
  }
  const float bv = bC[colBase + nl];

  const int ldrow = tid >> 4;
  const int ldcol = (tid & 15) * 8;
  const unsigned ldsStage =
      (unsigned)(size_t)&lds[(size_t)ldrow * LDSROW + ldcol];

  const int NT = N_EDGES / 16;
  const int step = gridDim.x;
  int t = blockIdx.x;
  int cur = 0;

  if (t < NT)
    async_stage32(ldsStage, e + (size_t)(t * 16 + ldrow) * DIM + ldcol);
  wait_async0();
  __syncthreads();

  for (; t < NT; t += step) {
    const int tn = t + step;
    if (tn < NT)
      async_stage32(ldsStage + (cur ^ 1) * TILE_BYTES,
                    e + (size_t)(tn * 16 + ldrow) * DIM + ldcol);

    v8f acc0 = {bv, bv, bv, bv, bv, bv, bv, bv};
    v8f acc1 = {0.f, 0.f, 0.f, 0.f, 0.f, 0.f, 0.f, 0.f};
    const float* arow = &lds[cur * TILE_FLOATS + nl * LDSROW + kk];
#pragma unroll
    for (int ks = 0; ks < 32; ks += 2) {
      v2f a0 = *(const v2f*)(arow + 4 * ks);
      acc0 = wmma_f32_k4(a0, Bfrag[ks], acc0);
      v2f a1 = *(const v2f*)(arow + 4 * (ks + 1));
      acc1 = wmma_f32_k4(a1, Bfrag[ks + 1], acc1);
    }

    const int rowBase = t * 16;
    const int r0  = (lane >> 4) * 8;
    const int col = colBase + nl;
#pragma unroll
    for (int j = 0; j < 8; ++j) {
      const int r    = r0 + j;
      const int eidx = rowBase + r;
      const int s = src[eidx];
      const int d = dst[eidx];
      float enew = (acc0[j] + acc1[j])
                 + Dh[(size_t)s * DIM + col] + Eh[(size_t)d * DIM + col];
      float sig = 1.0f / (1.0f + __expf(-enew));
      float msg = Bh[(size_t)s * DIM + col] * sig;
      atomicAdd(&sum_h[(size_t)d * DIM + col], msg);
      atomicAdd(&sum_s[(size_t)d * DIM + col], sig);
      // silu(enew) == enew * sigmoid(enew) == enew * sig
      e_out[(size_t)eidx * DIM + col] =
          lds[cur * TILE_FLOATS + r * LDSROW + col] + enew * sig;
    }

    wait_async0();
    __syncthreads();
    cur ^= 1;
  }
}

// ---------------------------------------------------------------------------
// Node finalize: h_out = h + silu(Ah + sum_h / (sum_s + eps))
// ---------------------------------------------------------------------------
__global__ void k_node_final(const float* __restrict__ h,
                             const float* __restrict__ Ah,
                             const float* __restrict__ ssh,
                             const float* __restrict__ sss,
                             float* __restrict__ h_out, int n) {
  int i = blockIdx.x * blockDim.x + threadIdx.x;
  if (i >= n) return;
  float hn  = Ah[i] + ssh[i] / (sss[i] + EPS_F);
  float sig = 1.0f / (1.0f + __expf(-hn));
  h_out[i] = h[i] + hn * sig;
}

// ---------------------------------------------------------------------------
extern "C" void kernel_launch(void* const* d_in, const int* in_sizes, int n_in,
                              void* d_out, int out_size, void* d_ws, size_t ws_size,
                              hipStream_t stream) {
  const float* h  = (const float*)d_in[0];
  const float* e  = (const float*)d_in[1];
  const int* src  = (const int*)d_in[2];
  const int* dst  = (const int*)d_in[3];
  const float* WA = (const float*)d_in[4];  const float* bA = (const float*)d_in[5];
  const float* WB = (const float*)d_in[6];  const float* bB = (const float*)d_in[7];
  const float* WC = (const float*)d_in[8];  const float* bC = (const float*)d_in[9];
  const float* WD = (const float*)d_in[10]; const float* bD = (const float*)d_in[11];
  const float* WE = (const float*)d_in[12]; const float* bE = (const float*)d_in[13];

  float* ws = (float*)d_ws;
  const size_t ND = (size_t)N_NODES * DIM;
  float* Ah  = ws + 0 * ND;
  float* Bh  = ws + 1 * ND;
  float* Dh  = ws + 2 * ND;
  float* Eh  = ws + 3 * ND;
  float* ssh = ws + 4 * ND;
  float* sss = ws + 5 * ND;

  float* h_out = (float*)d_out;
  float* e_out = (float*)d_out + ND;

  {  // zero segment sums (2*ND floats)
    int n4 = (int)(2 * ND / 4);
    k_zero<<<(n4 + 255) / 256, 256, 0, stream>>>(ssh, n4);
  }
  k_node_gemm<<<dim3(4, 256), 256, 0, stream>>>(
      h, WA, bA, WB, bB, WD, bD, WE, bE, ws);
  k_edge<<<dim3(1024), 256, 0, stream>>>(
      e, src, dst, WC, bC, Bh, Dh, Eh, ssh, sss, e_out);
  k_node_final<<<(int)((ND + 255) / 256), 256, 0, stream>>>(
      h, Ah, ssh, sss, h_out, (int)ND);
}